// HeteroGraphormerStructuralBias_64330020159494
// MI455X (gfx1250) — compile-verified
//
#include <hip/hip_runtime.h>
#include <stdint.h>

typedef float v4f __attribute__((ext_vector_type(4)));
typedef int   v4i __attribute__((ext_vector_type(4)));

#define GN 4096
#define GT 8
#define GR 16
#define GH 8
#define GB 21

// CDNA5 async global->LDS copy (GVS mode): lds[idx] = g[idx].
// VDST vgpr = LDS byte address, VADDR vgpr = 32-bit byte offset, SADDR = 64-bit base.
// Tracked by ASYNCcnt; wave must s_wait_asynccnt before consuming.
__device__ __forceinline__ void async_ld_f32(const float* g, unsigned idx, float* lds) {
    unsigned lds_addr = (unsigned)(uintptr_t)(lds + idx);   // low 32 bits of generic LDS ptr = LDS offset
    unsigned voff     = idx * 4u;
    asm volatile("global_load_async_to_lds_b32 %0, %1, %2"
                 :
                 : "v"(lds_addr), "v"(voff), "s"(g)
                 : "memory");
}

__device__ __forceinline__ int bucket_dt(float dt) {
    // matches jnp: sign(dt)*log1p(|dt|+1e-6), clip [-5,5], /(10+1e-9)==10.0f in f32, floor(*20), clip [0,20]
    float sl = log1pf(fabsf(dt) + 1e-6f);
    sl = (dt > 0.0f) ? sl : (dt < 0.0f ? -sl : 0.0f);
    float cl   = fminf(fmaxf(sl, -5.0f), 5.0f);
    float norm = (cl + 5.0f) / 10.0f;
    int idx = (int)floorf(norm * (float)(GB - 1));
    return min(max(idx, 0), GB - 1);
}

// ---------------- CSR build (edge binning by source row) ----------------

__global__ __launch_bounds__(256) void zero_counts_kernel(int* __restrict__ counts) {
    counts[blockIdx.x * 256 + threadIdx.x] = 0;
}

__global__ __launch_bounds__(256) void count_edges_kernel(
    const int* __restrict__ edge_src, int* __restrict__ counts, int num_edges) {
    int e = (int)(blockIdx.x * 256u) + threadIdx.x;
    if (e < num_edges) atomicAdd(&counts[edge_src[e]], 1);
}

// Single block: exclusive scan of 4096 row counts -> offsets[4097], cursor copy.
__global__ __launch_bounds__(256) void scan_offsets_kernel(
    const int* __restrict__ counts, int* __restrict__ offsets, int* __restrict__ cursor) {
    __shared__ int part[256];
    const int tid  = threadIdx.x;
    const int base = tid * 16;
    int c[16];
    int s = 0;
#pragma unroll
    for (int k = 0; k < 16; ++k) { c[k] = counts[base + k]; s += c[k]; }
    part[tid] = s;
    __syncthreads();
    if (tid == 0) {
        int run = 0;
        for (int k = 0; k < 256; ++k) { int t = part[k]; part[k] = run; run += t; }
    }
    __syncthreads();
    int off = part[tid];
#pragma unroll
    for (int k = 0; k < 16; ++k) {
        offsets[base + k] = off;
        cursor [base + k] = off;
        off += c[k];
    }
    if (tid == 255) offsets[GN] = off;   // == num_edges
}

// Pack each edge as (rel<<12)|dst into its source row's CSR segment.
__global__ __launch_bounds__(256) void scatter_edges_kernel(
    const int* __restrict__ edge_src, const int* __restrict__ edge_dst,
    const int* __restrict__ edge_rel, int* __restrict__ cursor,
    int* __restrict__ edges_pk, int num_edges) {
    int e = (int)(blockIdx.x * 256u) + threadIdx.x;
    if (e < num_edges) {
        int pos = atomicAdd(&cursor[edge_src[e]], 1);
        edges_pk[pos] = (edge_rel[e] << 12) | edge_dst[e];
    }
}

// ---------------- Single-pass fill: base + temporal + edge adds (via LDS) ----------------
// One block = one row i, quarter of the j range. tid handles 4 consecutive j.
__global__ __launch_bounds__(256) void fill_bias_csr_kernel(
    const int*   __restrict__ token_type,
    const float* __restrict__ time_vec,
    const int*   __restrict__ seed_count_p,
    const float* __restrict__ typepair_bias,   // T*T*H = 512
    const float* __restrict__ temp_bias,       // B*H   = 168
    const float* __restrict__ adj_rel_bias,    // R*H   = 128
    const int*   __restrict__ row_off,         // 4097
    const int*   __restrict__ edges_pk,        // E, (rel<<12)|dst
    float*       __restrict__ out)
{
    __shared__ float tp [GT * GT * GH];    // 512 floats
    __shared__ float tb [GB * GH];         // 168 floats
    __shared__ float adj[GR * GH];         // 128 floats
    __shared__ float acc[GH * 1024];       // 32 KB edge accumulator, acc[h][jlocal]

    const int tid = threadIdx.x;

    // Stage tables via CDNA5 async copy path (no VGPR round trip)
    async_ld_f32(typepair_bias, tid,       tp);
    async_ld_f32(typepair_bias, tid + 256, tp);
    if (tid < GB * GH) async_ld_f32(temp_bias,    tid, tb);
    if (tid < GR * GH) async_ld_f32(adj_rel_bias, tid, adj);

    // Zero the edge accumulator while the async copies fly (32 floats/thread).
    {
        v4f z = {0.f, 0.f, 0.f, 0.f};
#pragma unroll
        for (int k = 0; k < 8; ++k)
            *(v4f*)&acc[tid * 32 + k * 4] = z;
    }
    __builtin_amdgcn_s_wait_asynccnt(0);
    __syncthreads();

    const int i     = (int)(blockIdx.x >> 2);
    const int chunk = (int)(blockIdx.x & 3u);
    const int j0    = (chunk << 10) + (tid << 2);
    const int jloc  = tid << 2;

    // Scatter this row's edges that land in our j chunk into LDS (duplicates accumulate).
    {
        const int rb = row_off[i];       // uniform -> scalar loads
        const int re = row_off[i + 1];
        for (int e = rb + tid; e < re; e += 256) {
            const int pk = edges_pk[e];
            const int d  = pk & 4095;
            if ((d >> 10) == chunk) {
                const int jl = d & 1023;
                const float* __restrict__ ar = &adj[(pk >> 12) * GH];
#pragma unroll
                for (int h = 0; h < GH; ++h)
                    atomicAdd(&acc[h * 1024 + jl], ar[h]);   // ds_add_f32
            }
        }
    }

    const int tti = token_type[i];                       // uniform -> scalar load
    const float* __restrict__ tprow = &tp[tti * (GT * GH)];

    const v4i ttj = *(const v4i*)(token_type + j0);      // 16B aligned
    const int o0 = ttj.x * GH, o1 = ttj.y * GH, o2 = ttj.z * GH, o3 = ttj.w * GH;

    const int  S      = seed_count_p[0];                 // uniform scalar
    const bool seeded = (i < S);
    int b0 = 0, b1 = 0, b2 = 0, b3 = 0;
    if (seeded) {
        const float tvi = time_vec[i];
        const v4f tvj = *(const v4f*)(time_vec + j0);
        b0 = bucket_dt(tvj.x - tvi) * GH;
        b1 = bucket_dt(tvj.y - tvi) * GH;
        b2 = bucket_dt(tvj.z - tvi) * GH;
        b3 = bucket_dt(tvj.w - tvi) * GH;
    }

    __syncthreads();   // accumulator complete

    const size_t base = ((size_t)i << 12) + (size_t)j0;
#pragma unroll
    for (int h = 0; h < GH; ++h) {
        v4f v;
        v.x = tprow[o0 + h];
        v.y = tprow[o1 + h];
        v.z = tprow[o2 + h];
        v.w = tprow[o3 + h];
        if (seeded) {
            v.x += tb[b0 + h];
            v.y += tb[b1 + h];
            v.z += tb[b2 + h];
            v.w += tb[b3 + h];
        }
        v += *(const v4f*)&acc[h * 1024 + jloc];   // conflict-free ds_load_b128
        // Output (512MB) >> L2 (192MB): stream it, don't cache it.
        __builtin_nontemporal_store(v, (v4f*)(out + ((size_t)h << 24) + base));
    }
}

// ---------------- Fallback path (used only if d_ws is too small) ----------------

__global__ __launch_bounds__(256) void fill_bias_kernel(
    const int*   __restrict__ token_type,
    const float* __restrict__ time_vec,
    const int*   __restrict__ seed_count_p,
    const float* __restrict__ typepair_bias,
    const float* __restrict__ temp_bias,
    float*       __restrict__ out)
{
    __shared__ float tp[GT * GT * GH];
    __shared__ float tb[GB * GH];
    const int tid = threadIdx.x;
    async_ld_f32(typepair_bias, tid,       tp);
    async_ld_f32(typepair_bias, tid + 256, tp);
    if (tid < GB * GH) async_ld_f32(temp_bias, tid, tb);
    __builtin_amdgcn_s_wait_asynccnt(0);
    __syncthreads();

    const int i  = (int)(blockIdx.x >> 2);
    const int j0 = (int)((blockIdx.x & 3u) << 10) + (tid << 2);
    const int tti = token_type[i];
    const float* __restrict__ tprow = &tp[tti * (GT * GH)];
    const v4i ttj = *(const v4i*)(token_type + j0);
    const int o0 = ttj.x * GH, o1 = ttj.y * GH, o2 = ttj.z * GH, o3 = ttj.w * GH;
    const int  S      = seed_count_p[0];
    const bool seeded = (i < S);
    int b0 = 0, b1 = 0, b2 = 0, b3 = 0;
    if (seeded) {
        const float tvi = time_vec[i];
        const v4f tvj = *(const v4f*)(time_vec + j0);
        b0 = bucket_dt(tvj.x - tvi) * GH;
        b1 = bucket_dt(tvj.y - tvi) * GH;
        b2 = bucket_dt(tvj.z - tvi) * GH;
        b3 = bucket_dt(tvj.w - tvi) * GH;
    }
    const size_t base = ((size_t)i << 12) + (size_t)j0;
#pragma unroll
    for (int h = 0; h < GH; ++h) {
        v4f v;
        v.x = tprow[o0 + h]; v.y = tprow[o1 + h];
        v.z = tprow[o2 + h]; v.w = tprow[o3 + h];
        if (seeded) {
            v.x += tb[b0 + h]; v.y += tb[b1 + h];
            v.z += tb[b2 + h]; v.w += tb[b3 + h];
        }
        __builtin_nontemporal_store(v, (v4f*)(out + ((size_t)h << 24) + base));
    }
}

__global__ __launch_bounds__(256) void edge_scatter_kernel(
    const int*   __restrict__ edge_src,
    const int*   __restrict__ edge_dst,
    const int*   __restrict__ edge_rel,
    const float* __restrict__ adj_rel_bias,
    float*       __restrict__ out,
    int num_edges)
{
    __shared__ float a[GR * GH];
    const int tid = threadIdx.x;
    if (tid < GR * GH) async_ld_f32(adj_rel_bias, tid, a);
    __builtin_amdgcn_s_wait_asynccnt(0);
    __syncthreads();
    const int e = (int)(blockIdx.x * 256u) + tid;
    if (e >= num_edges) return;
    const int s = edge_src[e];
    const int d = edge_dst[e];
    const int r = edge_rel[e];
    const float* __restrict__ ar = &a[r * GH];
    float* p = out + ((size_t)s << 12) + (size_t)d;
#pragma unroll
    for (int h = 0; h < GH; ++h)
        unsafeAtomicAdd(p + ((size_t)h << 24), ar[h]);   // global_atomic_add_f32, no return
}

extern "C" void kernel_launch(void* const* d_in, const int* in_sizes, int n_in,
                              void* d_out, int out_size, void* d_ws, size_t ws_size,
                              hipStream_t stream) {
    const int*   token_type    = (const int*)  d_in[0];
    const int*   edge_src      = (const int*)  d_in[1];
    const int*   edge_dst      = (const int*)  d_in[2];
    const int*   edge_rel      = (const int*)  d_in[3];
    const float* time_vec      = (const float*)d_in[4];
    const int*   seed_count    = (const int*)  d_in[5];
    const float* adj_rel_bias  = (const float*)d_in[6];
    const float* typepair_bias = (const float*)d_in[7];
    const float* temp_bias     = (const float*)d_in[8];
    float* out = (float*)d_out;

    const int num_edges = in_sizes[1];

    // Workspace layout (ints): counts[4096] | offsets[4097] | cursor[4096] | edges_pk[E]
    const size_t ws_needed = (size_t)(GN + (GN + 1) + GN + num_edges) * sizeof(int);

    if (ws_size >= ws_needed) {
        int* w        = (int*)d_ws;
        int* counts   = w;
        int* offsets  = w + GN;
        int* cursor   = w + GN + (GN + 1);
        int* edges_pk = w + GN + (GN + 1) + GN;

        zero_counts_kernel<<<GN / 256, 256, 0, stream>>>(counts);
        count_edges_kernel<<<(num_edges + 255) / 256, 256, 0, stream>>>(edge_src, counts, num_edges);
        scan_offsets_kernel<<<1, 256, 0, stream>>>(counts, offsets, cursor);
        scatter_edges_kernel<<<(num_edges + 255) / 256, 256, 0, stream>>>(
            edge_src, edge_dst, edge_rel, cursor, edges_pk, num_edges);
        // Single streaming pass over the 512MB output: no global atomics.
        fill_bias_csr_kernel<<<GN * 4, 256, 0, stream>>>(
            token_type, time_vec, seed_count, typepair_bias, temp_bias,
            adj_rel_bias, offsets, edges_pk, out);
    } else {
        // Fallback: dense fill + global-atomic edge scatter.
        fill_bias_kernel<<<GN * 4, 256, 0, stream>>>(token_type, time_vec, seed_count,
                                                     typepair_bias, temp_bias, out);
        edge_scatter_kernel<<<(num_edges + 255) / 256, 256, 0, stream>>>(
            edge_src, edge_dst, edge_rel, adj_rel_bias, out, num_edges);
    }
}